// GCNClassifier_60284160967392
// MI455X (gfx1250) — compile-verified
//
#include <hip/hip_runtime.h>

// GCN classifier for MI455X (gfx1250, wave32).
// - GEMMs via V_WMMA_F32_16X16X4_F32 (full fp32, one wave per 16x128 strip)
// - W staged once per workgroup into LDS (k-pair-interleaved float2, padded
//   stride => conflict-free 64-bank b64 reads), cutting L2 broadcast traffic 8x
// - Edge aggregation via float4 gathers (+ global_prefetch of next source row)
//   and native global_atomic_add_f32 scatters; h fits in the 192MB L2
#define HDIM 128
#define WSTRIDE 144   // float2 units; 144*8=1152B == 128 (mod 256) => odd k-pairs
                      // map to banks 32..63 while even k-pairs use banks 0..31

typedef __attribute__((ext_vector_type(2))) float v2f;
typedef __attribute__((ext_vector_type(8))) float v8f;

__global__ void k_zero(float* __restrict__ p, long n) {
  long i = (long)blockIdx.x * blockDim.x + threadIdx.x;
  long stride = (long)gridDim.x * blockDim.x;
  for (; i < n; i += stride) p[i] = 0.0f;
}

__global__ void k_deg(const int* __restrict__ dst, const float* __restrict__ ew,
                      float* __restrict__ deg, int E) {
  int i = blockIdx.x * blockDim.x + threadIdx.x;
  int stride = gridDim.x * blockDim.x;
  for (; i < E; i += stride) unsafeAtomicAdd(&deg[dst[i]], ew[i]);
}

__global__ void k_dinv(const float* __restrict__ deg, float* __restrict__ dinv, int N) {
  int i = blockIdx.x * blockDim.x + threadIdx.x;
  if (i < N) {
    float d = deg[i] + 1.0f;          // +1 for self-loop
    dinv[i] = (d > 0.0f) ? rsqrtf(d) : 0.0f;
  }
}

// h[N,128] = X[N,128] @ W[128,128], one wave per 16-row strip (all 128 cols).
// A frag (16x4 f32): lane L holds row (L&15), K pair (L>>4)*2 .. +1.
// B frag (4x16 f32): lane L holds col (L&15), same K pair -> one LDS float2.
// C/D (16x16 f32): VGPR r -> M = r + (L>>4)*8, N = L&15.
__global__ __launch_bounds__(256) void k_gemm128(const float* __restrict__ X,
                                                 const float* __restrict__ W,
                                                 float* __restrict__ Hout, int N) {
  __shared__ float2 Wlds[64 * WSTRIDE];

  // Stage W k-pair-interleaved: Wlds[p*WSTRIDE + c] = {W[2p][c], W[2p+1][c]}.
  // For fixed p the 128 c-lanes read two contiguous rows -> coalesced b32 pairs.
  for (int idx = threadIdx.x; idx < 64 * HDIM; idx += 256) {
    int p = idx >> 7;
    int c = idx & (HDIM - 1);
    const float* wr = W + (long)(p << 1) * HDIM + c;
    Wlds[p * WSTRIDE + c] = make_float2(wr[0], wr[HDIM]);
  }
  __syncthreads();

  int lane = threadIdx.x & 31;
  int wave = blockIdx.x * (blockDim.x >> 5) + (threadIdx.x >> 5);
  int row0 = wave * 16;
  if (row0 + 16 > N) return;          // N is a multiple of 16 in this workload

  int m  = lane & 15;                 // row (A) / col (B) within tile
  int kh = (lane >> 4) << 1;          // K sub-pair: 0 or 2

  v8f acc[8] = {};
  const float* xrow = X + (long)(row0 + m) * HDIM;

  for (int k = 0; k < HDIM; k += 4) {
    v2f a;
    a.x = xrow[k + kh];
    a.y = xrow[k + kh + 1];
    // pair index for this lane: k/2 (lanes 0-15) or k/2+1 (lanes 16-31)
    const float2* wp = Wlds + ((k >> 1) + (lane >> 4)) * WSTRIDE;
#pragma unroll
    for (int j = 0; j < 8; ++j) {
      float2 bb = wp[j * 16 + m];     // ds_load_b64, conflict-free
      v2f b; b.x = bb.x; b.y = bb.y;
      acc[j] = __builtin_amdgcn_wmma_f32_16x16x4_f32(
          /*neg_a=*/false, a, /*neg_b=*/false, b,
          /*c_mod=*/(short)0, acc[j], /*reuse_a=*/false, /*reuse_b=*/false);
    }
  }

  int rOff = (lane >> 4) * 8;
#pragma unroll
  for (int j = 0; j < 8; ++j) {
#pragma unroll
    for (int r = 0; r < 8; ++r) {
      Hout[(long)(row0 + rOff + r) * HDIM + j * 16 + m] = acc[j][r];
    }
  }
}

// agg[dst] += h[src] * (dinv[src]*ew*dinv[dst]); one wave per edge, 4 floats/lane.
// Prefetch next edge's source row to overlap L2 gather latency with atomics.
__global__ __launch_bounds__(256) void k_edge(const float* __restrict__ h,
                                              const int* __restrict__ src,
                                              const int* __restrict__ dst,
                                              const float* __restrict__ ew,
                                              const float* __restrict__ dinv,
                                              float* __restrict__ agg, int E) {
  int lane = threadIdx.x & 31;
  int wid  = blockIdx.x * (blockDim.x >> 5) + (threadIdx.x >> 5);
  int nw   = gridDim.x * (blockDim.x >> 5);
  for (int e = wid; e < E; e += nw) {
    int s = src[e], d = dst[e];
    int eNext = e + nw;
    if (eNext < E) {
      int sn = src[eNext];
      __builtin_prefetch(h + (long)sn * HDIM + lane * 4, 0, 1);  // global_prefetch
    }
    float nrm = dinv[s] * ew[e] * dinv[d];
    float4 v = ((const float4*)(h + (long)s * HDIM))[lane];      // b128 gather (L2 hit)
    float* ad = agg + (long)d * HDIM + lane * 4;
    unsafeAtomicAdd(ad + 0, v.x * nrm);
    unsafeAtomicAdd(ad + 1, v.y * nrm);
    unsafeAtomicAdd(ad + 2, v.z * nrm);
    unsafeAtomicAdd(ad + 3, v.w * nrm);
  }
}

// y = relu(agg + h * dinv^2 + bias), in place over agg buffer.
__global__ void k_finalize(float* __restrict__ y, const float* __restrict__ h,
                           const float* __restrict__ dinv,
                           const float* __restrict__ bias, int N) {
  long total = (long)N * HDIM;
  long i = (long)blockIdx.x * blockDim.x + threadIdx.x;
  long stride = (long)gridDim.x * blockDim.x;
  for (; i < total; i += stride) {
    int n = (int)(i >> 7);
    int f = (int)(i & (HDIM - 1));
    float di = dinv[n];
    float v = y[i] + h[i] * di * di + bias[f];
    y[i] = fmaxf(v, 0.0f);
  }
}

// pooled[batch[n]] += y[n]; cnt[batch[n]] += 1. One wave per node.
__global__ __launch_bounds__(256) void k_pool(const float* __restrict__ y,
                                              const int* __restrict__ batch,
                                              float* __restrict__ pooled,
                                              float* __restrict__ cnt, int N) {
  int lane = threadIdx.x & 31;
  int wid  = blockIdx.x * (blockDim.x >> 5) + (threadIdx.x >> 5);
  int nw   = gridDim.x * (blockDim.x >> 5);
  for (int n = wid; n < N; n += nw) {
    int b = batch[n];
    float4 v = ((const float4*)(y + (long)n * HDIM))[lane];
    float* p = pooled + (long)b * HDIM + lane * 4;
    unsafeAtomicAdd(p + 0, v.x);
    unsafeAtomicAdd(p + 1, v.y);
    unsafeAtomicAdd(p + 2, v.z);
    unsafeAtomicAdd(p + 3, v.w);
    if (lane == 0) unsafeAtomicAdd(&cnt[b], 1.0f);
  }
}

// out[g,c] = (pooled[g]/max(cnt,1)) @ fc_w + fc_b ; G=64, C=2 -> trivial.
__global__ void k_fc(const float* __restrict__ pooled, const float* __restrict__ cnt,
                     const float* __restrict__ fcw, const float* __restrict__ fcb,
                     float* __restrict__ out, int G) {
  int g = blockIdx.x * blockDim.x + threadIdx.x;
  if (g >= G) return;
  float inv = 1.0f / fmaxf(cnt[g], 1.0f);
  float a0 = 0.0f, a1 = 0.0f;
  const float* p = pooled + (long)g * HDIM;
  for (int f = 0; f < HDIM; ++f) {
    float v = p[f] * inv;
    a0 += v * fcw[f * 2 + 0];
    a1 += v * fcw[f * 2 + 1];
  }
  out[g * 2 + 0] = a0 + fcb[0];
  out[g * 2 + 1] = a1 + fcb[1];
}

extern "C" void kernel_launch(void* const* d_in, const int* in_sizes, int n_in,
                              void* d_out, int out_size, void* d_ws, size_t ws_size,
                              hipStream_t stream) {
  // setup_inputs order: x, edge_index, edge_attr, batch, W1,b1, W2,b2, W3,b3, fc_w,fc_b
  const float* x     = (const float*)d_in[0];
  const int*   ei    = (const int*)d_in[1];
  const float* ea    = (const float*)d_in[2];
  const int*   batch = (const int*)d_in[3];
  const float* Ws[3] = {(const float*)d_in[4], (const float*)d_in[6], (const float*)d_in[8]};
  const float* Bs[3] = {(const float*)d_in[5], (const float*)d_in[7], (const float*)d_in[9]};
  const float* fcw   = (const float*)d_in[10];
  const float* fcb   = (const float*)d_in[11];

  const int N = in_sizes[0] / HDIM;
  const int E = in_sizes[2];
  const int G = out_size / 2;
  const int* src = ei;
  const int* dst = ei + E;

  float* ws     = (float*)d_ws;
  float* deg    = ws;                 ws += N;
  float* dinv   = ws;                 ws += N;
  float* Hbuf   = ws;                 ws += (size_t)N * HDIM;
  float* Ybuf   = ws;                 ws += (size_t)N * HDIM;
  float* pooled = ws;                 ws += (size_t)G * HDIM;
  float* cnt    = ws;                 ws += G;  // contiguous after pooled

  const long NH = (long)N * HDIM;

  // Degree + normalization (once, reused by all 3 layers).
  k_zero<<<256, 256, 0, stream>>>(deg, N);
  k_deg<<<1024, 256, 0, stream>>>(dst, ea, deg, E);
  k_dinv<<<(N + 255) / 256, 256, 0, stream>>>(deg, dinv, N);

  const int tiles = N / 16;                       // N is a multiple of 16
  const int gemmBlocks = (tiles + 7) / 8;         // 8 waves per 256-thread block

  const float* inp = x;
  for (int L = 0; L < 3; ++L) {
    k_gemm128<<<gemmBlocks, 256, 0, stream>>>(inp, Ws[L], Hbuf, N);
    k_zero<<<1024, 256, 0, stream>>>(Ybuf, NH);   // input (Ybuf) already consumed by GEMM
    k_edge<<<4096, 256, 0, stream>>>(Hbuf, src, dst, ea, dinv, Ybuf, E);
    k_finalize<<<2048, 256, 0, stream>>>(Ybuf, Hbuf, dinv, Bs[L], N);
    inp = Ybuf;
  }

  k_zero<<<64, 256, 0, stream>>>(pooled, (long)G * HDIM + G);  // pooled + cnt
  k_pool<<<2048, 256, 0, stream>>>(Ybuf, batch, pooled, cnt, N);
  k_fc<<<(G + 63) / 64, 64, 0, stream>>>(pooled, cnt, fcw, fcb, (float*)d_out, G);
}